// mah_distance_loss_35579509080732
// MI455X (gfx1250) — compile-verified
//
#include <hip/hip_runtime.h>
#include <hip/hip_bf16.h>

typedef __attribute__((ext_vector_type(2))) float v2f;
typedef __attribute__((ext_vector_type(8))) float v8f;
typedef __attribute__((ext_vector_type(4))) unsigned int v4u;
typedef __attribute__((ext_vector_type(8))) int v8i;
typedef __attribute__((ext_vector_type(4))) int v4i;

static constexpr int NN   = 1024;
static constexpr int DD   = 128;
static constexpr int FF   = 256;
static constexpr int MAXE = (NN * (NN - 1)) / 2 + 256; // padded upper bound
static constexpr int NCHUNK    = 64;                   // K-chunks for Gram pass
static constexpr int NTILES    = 136;                  // 16*17/2 upper-tri tiles
static constexpr int MD_BLOCKS = 256;                  // 256 blk x 8 waves = 2048 partials
static constexpr int NPART     = MD_BLOCKS * 8;

// ---- workspace layout (bytes) ----
static constexpr size_t OFF_HDR  = 0;        // int[16]   hdr[0]=E hdr[1]=Epad
static constexpr size_t OFF_SCAL = 64;       // float[16] [0]=threshold [1]=c0
static constexpr size_t OFF_MU   = 128;      // float[256]
static constexpr size_t OFF_V    = 1152;     // float[256]  v = S^-1 mu
static constexpr size_t OFF_CNTI = 2176;     // int[1024]
static constexpr size_t OFF_CNTJ = 6272;     // int[1024]
static constexpr size_t OFF_PART = 10368;    // float[NPART] -> ends 18560
static constexpr size_t OFF_G    = 32768;    // float[256*256] X^T X, later S^-1 compact
static constexpr size_t OFF_AUG  = 294912;   // float[256*512] [S|I] -> [I|S^-1]
static constexpr size_t OFF_EDGE = 819200;   // int2[MAXE]  (16B aligned)
static constexpr size_t OFF_EMBZ = OFF_EDGE + (size_t)MAXE * 8;  // float[1025*128]
static constexpr int    ZERO_WORDS = (int)(OFF_AUG / 4);         // zero [0, OFF_AUG)

__global__ void k_init(float* ws_f) {
  for (int i = blockIdx.x * blockDim.x + threadIdx.x; i < ZERO_WORDS;
       i += gridDim.x * blockDim.x)
    ws_f[i] = 0.0f;
}

// embz = emb with one extra all-zero row at index NN (sentinel for padding).
__global__ void k_embz(const float* __restrict__ adj, float* __restrict__ embz) {
  int i = blockIdx.x * blockDim.x + threadIdx.x;
  if (i < (NN + 1) * DD) embz[i] = (i < NN * DD) ? adj[i] : 0.0f;
}

__global__ void k_extract(const int* __restrict__ d, int* hdr, int2* edges,
                          int* cnt_i, int* cnt_j) {
  int idx = blockIdx.x * blockDim.x + threadIdx.x;
  if (idx >= NN * NN) return;
  int r = idx >> 10, c = idx & (NN - 1);
  if (r < c && d[idx] != 0) {
    int pos = atomicAdd(&hdr[0], 1);
    edges[pos] = make_int2(r, c);
    atomicAdd(&cnt_i[r], 1);
    atomicAdd(&cnt_j[c], 1);
  }
}

// Pad edge list to a multiple of 256 with sentinel edges -> zero row of embz.
__global__ void k_pad(int* hdr, int2* edges) {
  int E = hdr[0];
  int Epad = (E + 255) & ~255;
  int t = threadIdx.x;
  if (E + t < Epad) edges[E + t] = make_int2(NN, NN);
  if (t == 0) hdr[1] = Epad;
}

// mu[f] = (sum_e X[e][f]) / E  via degree-weighted node sums.
__global__ void k_mu(const float* __restrict__ emb, const int* cnt_i,
                     const int* cnt_j, const int* hdr, float* mu) {
  int f = threadIdx.x;
  float s = 0.0f;
  if (f < DD) {
    for (int n = 0; n < NN; ++n) s += (float)cnt_i[n] * emb[n * DD + f];
  } else {
    int ff = f - DD;
    for (int n = 0; n < NN; ++n) s += (float)cnt_j[n] * emb[n * DD + ff];
  }
  mu[f] = s / (float)hdr[0];
}

// G = X^T X via fp32 WMMA 16x16x4, upper-triangular tiles only (G symmetric).
// grid = (NTILES, NCHUNK), block = 1 wave. Branch-free hot loop (padded edges
// hit the zero sentinel row).
__global__ void k_gram(const float* __restrict__ embz,
                       const int2* __restrict__ edges,
                       const int* __restrict__ hdr, float* G) {
  // decode linear upper-tri tile index -> (mt, nt), mt <= nt
  int t = blockIdx.x, mt = 0, rl = 16;
  while (t >= rl) { t -= rl; --rl; ++mt; }
  const int nt = mt + t;
  const int m0 = mt << 4, n0 = nt << 4;
  const int chunk = blockIdx.y;
  const int lane = threadIdx.x, half = lane >> 4, l16 = lane & 15;
  const int Epad = hdr[1];
  const bool mI = (m0 < DD), nI = (n0 < DD);
  const int mf = (m0 & (DD - 1)) + l16, nf = (n0 & (DD - 1)) + l16;
  v8f acc = {};
  for (int e0 = chunk * 32; e0 < Epad; e0 += NCHUNK * 32) {
#pragma unroll
    for (int u = 0; u < 8; ++u) {
      const int eb = e0 + u * 4 + half * 2;          // this half's 2 edges
      int4 ee = *(const int4*)(edges + eb);          // (i0,j0,i1,j1)
      v2f a, b;
      a[0] = embz[(mI ? ee.x : ee.y) * DD + mf];     // A(M=l16, K=2*half+v)
      a[1] = embz[(mI ? ee.z : ee.w) * DD + mf];
      b[0] = embz[(nI ? ee.x : ee.y) * DD + nf];     // B(K=2*half+v, N=l16)
      b[1] = embz[(nI ? ee.z : ee.w) * DD + nf];
      acc = __builtin_amdgcn_wmma_f32_16x16x4_f32(false, a, false, b,
                                                  (short)0, acc, false, false);
    }
  }
#pragma unroll
  for (int v = 0; v < 8; ++v) {
    int row = m0 + v + half * 8, col = n0 + l16;     // C: M = v + 8*half, N = l16
    atomicAdd(&G[row * FF + col], acc[v]);
  }
}

// Aug = [ (G - E*mu*mu^T)/(E-1) | I ]  with symmetric mirror of G tiles.
__global__ void k_fin_s(const float* __restrict__ G, const float* __restrict__ mu,
                        const int* hdr, float* Aug) {
  int r = blockIdx.x, c = threadIdx.x;
  float Ef = (float)hdr[0];
  float g = ((r >> 4) <= (c >> 4)) ? G[r * FF + c] : G[c * FF + r];
  Aug[r * 512 + c] = (g - Ef * mu[r] * mu[c]) / (Ef - 1.0f);
  Aug[r * 512 + FF + c] = (r == c) ? 1.0f : 0.0f;
}

// Gauss-Jordan, single 256-thread block. S is SPD -> diagonal pivots safe.
__global__ void k_gj(float* Aug) {
  __shared__ float prow[512];
  const int t = threadIdx.x;
  for (int k = 0; k < FF; ++k) {
    float ip = 1.0f / Aug[k * 512 + k];
    __syncthreads();
    for (int c = t; c < 512; c += 256) {
      float val = Aug[k * 512 + c] * ip;
      Aug[k * 512 + c] = val;
      prow[c] = val;
    }
    __syncthreads();
    if (t != k) {
      float fac = Aug[t * 512 + k];
      for (int c = 0; c < 512; ++c) Aug[t * 512 + c] -= fac * prow[c];
    }
    __syncthreads();
  }
}

// Compact S^-1 (right half of Aug) into contiguous Tm[256][256] (reuses G).
__global__ void k_csinv(const float* __restrict__ Aug, float* __restrict__ Tm) {
  int r = blockIdx.x, c = threadIdx.x;
  Tm[r * FF + c] = Aug[r * 512 + FF + c];
}

// v = S^-1 mu ; c0 = mu.v ; threshold (Wilson-Hilferty in double, like numpy).
__global__ void k_thresh(const int* hdr, const float* __restrict__ Tm,
                         const float* __restrict__ mu, float* vvec, float* scal) {
  __shared__ float red[256];
  const int f = threadIdx.x;
  float s = 0.0f;
  for (int c = 0; c < FF; ++c) s += Tm[f * FF + c] * mu[c];
  vvec[f] = s;
  red[f] = s * mu[f];
  __syncthreads();
  for (int off = 128; off > 0; off >>= 1) {
    if (f < off) red[f] += red[f + off];
    __syncthreads();
  }
  if (f == 0) {
    scal[1] = red[0];
    double E  = (double)hdr[0];
    double df = E - (256.0 + E + 1.0) / 2.0;
    double cc = 2.0 / (9.0 * df);
    double z  = -1.6448536269514722;
    double tt = 1.0 - cc + z * sqrt(cc);
    scal[0] = (float)(df * tt * tt * tt);
  }
}

// md pass: 8 waves/block, each wave owns one 16-edge group per iteration.
// T = S^-1 staged into LDS in 256x64 column blocks via the Tensor Data Mover
// (one wave issues tensor_load_to_lds; all 8 waves consume after the barrier).
__global__ void __launch_bounds__(256)
k_md(const float* __restrict__ embz, const int2* __restrict__ edges,
     const int* __restrict__ hdr, const float* __restrict__ Tm,
     const float* __restrict__ vvec, const float* __restrict__ scal,
     float* partial) {
  __shared__ float Ts[FF * 64];          // 64 KB column block of T (row-major 64 wide)
  __shared__ float redq[8][32][8];
  __shared__ float redp[8][32];
  const int tid = threadIdx.x;
  const int wid = tid >> 5, lane = tid & 31;
  const int half = lane >> 4, l16 = lane & 15;
  const int E = hdr[0], Epad = hdr[1];
  const int nGroups = Epad >> 4;         // multiple of 16 -> multiple of 8
  const float thr = scal[0], c0 = scal[1];
  float psum = 0.0f;

  for (int g0i = blockIdx.x * 8; g0i < nGroups; g0i += gridDim.x * 8) {
    const int e0 = (g0i + wid) * 16;
    // lane's A-row edge (M = l16), invariant over the whole group
    const int2 ij = edges[e0 + l16];
    const float* rowi = embz + ij.x * DD;
    const float* rowj = embz + ij.y * DD;
    // the 8 C-rows this lane folds for the diagonal: M = v + 8*half
    int baseI[8], baseJ[8];
#pragma unroll
    for (int v = 0; v < 8; ++v) {
      int2 ijf = edges[e0 + v + half * 8];
      baseI[v] = ijf.x * DD;
      baseJ[v] = ijf.y * DD;
    }
    float pq[8] = {0, 0, 0, 0, 0, 0, 0, 0};

    for (int cb = 0; cb < 4; ++cb) {     // 64-column block of T
      __syncthreads();                   // everyone done reading previous Ts
#if __has_builtin(__builtin_amdgcn_tensor_load_to_lds) && \
    __has_builtin(__builtin_amdgcn_s_wait_tensorcnt)
      if (wid == 0) {
        // D# group0: count=1 | lds_addr | global tile addr | type=2 (ISA 8.3)
        unsigned long long ga =
            (unsigned long long)(size_t)(Tm + (size_t)cb * 64);
        unsigned int ldsb = (unsigned int)(size_t)Ts;   // LDS byte offset
        v4u g0 = { 1u, ldsb, (unsigned int)ga,
                   (unsigned int)((ga >> 32) & 0x1ffffffu) | (2u << 30) };
        // D# group1 (ISA 8.4): data_size=2(4B); tensor_dim0=256; tensor_dim1=256;
        // tile_dim0=64; tile_dim1=256; tensor_dim0_stride=256 (elements)
        v8i g1 = { (int)0x20000,         // wg_mask=0, data_size=4B
                   (int)(256u << 16),    // tensor_dim0[15:0]=256
                   (int)(256u << 16),    // tensor_dim0 hi=0 | tensor_dim1 lo=256
                   (int)(64u << 16),     // tensor_dim1 hi=0 | tile_dim0=64
                   256,                  // tile_dim1=256 | tile_dim2=0
                   256, 0, 0 };          // tensor_dim0_stride=256
        v4i z4 = { 0, 0, 0, 0 };
        v8i z8 = { 0, 0, 0, 0, 0, 0, 0, 0 };
        __builtin_amdgcn_tensor_load_to_lds(g0, g1, z4, z4, z8, 0);
        __builtin_amdgcn_s_wait_tensorcnt(0);
      }
#else
      for (int idx = tid; idx < FF * 64; idx += 256) {
        int r = idx >> 6, c = idx & 63;
        Ts[idx] = Tm[r * FF + cb * 64 + c];
      }
#endif
      __syncthreads();                   // Ts block visible to all waves
      for (int nt = 0; nt < 4; ++nt) {   // 16-column WMMA subtile
        const float* tcol = Ts + nt * 16 + l16;
        v8f acc = {};
#pragma unroll 4
        for (int ks = 0; ks < 32; ++ks) {            // K = feats 0..127 (rowi)
          const int fA = ks * 4 + half * 2;
          float2 av = *(const float2*)(rowi + fA);
          v2f a, b;
          a[0] = av.x; a[1] = av.y;
          b[0] = tcol[fA * 64];
          b[1] = tcol[(fA + 1) * 64];
          acc = __builtin_amdgcn_wmma_f32_16x16x4_f32(false, a, false, b,
                                                      (short)0, acc, false, false);
        }
#pragma unroll 4
        for (int ks = 0; ks < 32; ++ks) {            // K = feats 128..255 (rowj)
          const int fA = ks * 4 + half * 2;
          float2 av = *(const float2*)(rowj + fA);
          v2f a, b;
          a[0] = av.x; a[1] = av.y;
          b[0] = tcol[(fA + DD) * 64];
          b[1] = tcol[(fA + DD + 1) * 64];
          acc = __builtin_amdgcn_wmma_f32_16x16x4_f32(false, a, false, b,
                                                      (short)0, acc, false, false);
        }
        const int colf = ((cb * 64 + nt * 16 + l16) & (DD - 1));
        const bool colI = (cb < 2);
#pragma unroll
        for (int v = 0; v < 8; ++v)                  // diag: Y .* X fold
          pq[v] += acc[v] * embz[(colI ? baseI[v] : baseJ[v]) + colf];
      }
    }
#pragma unroll
    for (int v = 0; v < 8; ++v) redq[wid][lane][v] = pq[v];
    __syncthreads();
    if (lane < 16) {
      int e = e0 + lane;
      if (e < E) {
        int h = lane >> 3, v = lane & 7;
        float q = 0.0f;
        for (int t = 0; t < 16; ++t) q += redq[wid][h * 16 + t][v];
        int2 ije = edges[e];
        const float* ri = embz + ije.x * DD;
        const float* rj = embz + ije.y * DD;
        float dv = 0.0f;
        for (int f = 0; f < DD; ++f) dv += ri[f] * vvec[f];
        for (int f = 0; f < DD; ++f) dv += rj[f] * vvec[DD + f];
        float r = (q - 2.0f * dv + c0) - thr;
        psum += (r > 0.0f) ? r : 0.0f;
      }
    }
    __syncthreads();
  }
  redp[wid][lane] = psum;
  __syncthreads();
  if (lane == 0) {
    float s = 0.0f;
    for (int t = 0; t < 16; ++t) s += redp[wid][t];  // fixed order
    partial[blockIdx.x * 8 + wid] = s;
  }
}

__global__ void k_final(const int* hdr, const float* __restrict__ partial,
                        float* out) {
  __shared__ float red[256];
  const int t = threadIdx.x;
  float s = 0.0f;
  for (int i = t; i < NPART; i += 256) s += partial[i];   // fixed order
  red[t] = s;
  __syncthreads();
  for (int off = 128; off > 0; off >>= 1) {
    if (t < off) red[t] += red[t + off];
    __syncthreads();
  }
  if (t == 0) out[0] = red[0] / (float)hdr[0];
}

extern "C" void kernel_launch(void* const* d_in, const int* in_sizes, int n_in,
                              void* d_out, int out_size, void* d_ws, size_t ws_size,
                              hipStream_t stream) {
  const int*   d   = (const int*)d_in[0];     // [1,1024,1024] int32 adjacency
  const float* adj = (const float*)d_in[1];   // [1,1024,128] fp32 embeddings
  float* out = (float*)d_out;

  char* ws = (char*)d_ws;
  int*   hdr   = (int*)  (ws + OFF_HDR);
  float* scal  = (float*)(ws + OFF_SCAL);
  float* mu    = (float*)(ws + OFF_MU);
  float* vvec  = (float*)(ws + OFF_V);
  int*   cnt_i = (int*)  (ws + OFF_CNTI);
  int*   cnt_j = (int*)  (ws + OFF_CNTJ);
  float* part  = (float*)(ws + OFF_PART);
  float* G     = (float*)(ws + OFF_G);      // raw Gram, later compact S^-1
  float* Aug   = (float*)(ws + OFF_AUG);
  int2*  edges = (int2*) (ws + OFF_EDGE);
  float* embz  = (float*)(ws + OFF_EMBZ);

  k_init<<<288, 256, 0, stream>>>((float*)ws);
  k_embz<<<((NN + 1) * DD + 255) / 256, 256, 0, stream>>>(adj, embz);
  k_extract<<<(NN * NN) / 256, 256, 0, stream>>>(d, hdr, edges, cnt_i, cnt_j);
  k_pad<<<1, 256, 0, stream>>>(hdr, edges);
  k_mu<<<1, 256, 0, stream>>>(adj, cnt_i, cnt_j, hdr, mu);
  k_gram<<<dim3(NTILES, NCHUNK), 32, 0, stream>>>(embz, edges, hdr, G);
  k_fin_s<<<256, 256, 0, stream>>>(G, mu, hdr, Aug);
  k_gj<<<1, 256, 0, stream>>>(Aug);
  k_csinv<<<256, 256, 0, stream>>>(Aug, G);                 // G now holds S^-1
  k_thresh<<<1, 256, 0, stream>>>(hdr, G, mu, vvec, scal);
  k_md<<<MD_BLOCKS, 256, 0, stream>>>(embz, edges, hdr, G, vvec, scal, part);
  k_final<<<1, 256, 0, stream>>>(hdr, part, out);
}